// Quanv2d_21217138442864
// MI455X (gfx1250) — compile-verified
//
#include <hip/hip_runtime.h>

typedef float v2f __attribute__((ext_vector_type(2)));
typedef float v8f __attribute__((ext_vector_type(8)));

#define HW 192
#define OX 95
#define OY 95

__global__ __launch_bounds__(256)
void quanv_kernel(const float* __restrict__ X, const float* __restrict__ Wt,
                  float* __restrict__ out, int totalRows) {
    __shared__ float s_bmat[16 * 16];         // B = R^T : bmat[k*16+n] = R[n][k]
    __shared__ float s_phc[16], s_phs[16];    // CRZ diagonal phases
    __shared__ float s_re[8][32][18];         // staged encoded state (stride 18 pad)
    __shared__ float s_im[8][32][18];
    __shared__ float s_p[8][16][17];          // |amp|^2 per tile (stride 17 pad)

    const int tid  = threadIdx.x;
    const int lane = tid & 31;
    const int wv   = tid >> 5;

    // ---------------- block-shared precompute from weights ----------------
    const float w0 = Wt[0], w1 = Wt[1], w2 = Wt[2], w3 = Wt[3];
    float cq[4], sq[4];
#pragma unroll
    for (int q = 0; q < 4; ++q) __sincosf(0.5f * Wt[4 + q], &sq[q], &cq[q]);

    if (tid < 16) {
        int s = tid;
        int b0 = (s >> 3) & 1, b1 = (s >> 2) & 1, b2 = (s >> 1) & 1, b3 = s & 1;
        // CRZ(c=i, t=(i+1)%4): control=1 multiplies by e^{-i th/2} (t=0) or e^{+i th/2} (t=1)
        float phi = 0.5f * ((float)b0 * (2 * b1 - 1) * w0 + (float)b1 * (2 * b2 - 1) * w1 +
                            (float)b2 * (2 * b3 - 1) * w2 + (float)b3 * (2 * b0 - 1) * w3);
        float sp, cp; __sincosf(phi, &sp, &cp);
        s_phc[s] = cp; s_phs[s] = sp;
    }
    {   // R = kron of RY(w[4+q]); qubit 0 is MSB of the 4-bit basis index
        int k = tid >> 4, n = tid & 15;
        float v = 1.0f;
#pragma unroll
        for (int q = 0; q < 4; ++q) {
            int i = (n >> (3 - q)) & 1;          // output bit
            int j = (k >> (3 - q)) & 1;          // input bit
            float m = (i == j) ? cq[q] : ((i == 0) ? -sq[q] : sq[q]);
            v *= m;
        }
        s_bmat[k * 16 + n] = v;
    }
    __syncthreads();

    // ---------------- per-lane row: encode product state ----------------
    const int rowBase = (int)blockIdx.x * 256;
    int r = rowBase + tid;
    if (r >= totalRows) r = totalRows - 1;       // clamp; store predicated later

    unsigned rr  = (unsigned)r;
    unsigned b   = rr / (unsigned)(OX * OY);
    unsigned rem = rr % (unsigned)(OX * OY);
    unsigned px  = rem / (unsigned)OY;
    unsigned py  = rem % (unsigned)OY;

    const float* xp = X + (size_t)b * (HW * HW) + (size_t)(2u * px) * HW + 2u * py;
    float t[16];
#pragma unroll
    for (int kx = 0; kx < 4; ++kx) {
        const float* rp = xp + kx * HW;          // even float offset -> 8B aligned
        float2 a = *(const float2*)(rp);
        float2 c = *(const float2*)(rp + 2);
        t[kx * 4 + 0] = a.x; t[kx * 4 + 1] = a.y;
        t[kx * 4 + 2] = c.x; t[kx * 4 + 3] = c.y;
    }

    float vr[4][2], vi[4][2];
    const float is2 = 0.70710678118654752f;
#pragma unroll
    for (int q = 0; q < 4; ++q) { vr[q][0] = is2; vr[q][1] = is2; vi[q][0] = 0.f; vi[q][1] = 0.f; }

    auto applyRZ = [&](int q, float th) {
        float sh, ch; __sincosf(0.5f * th, &sh, &ch);
        float r0 = vr[q][0], i0 = vi[q][0], r1 = vr[q][1], i1 = vi[q][1];
        vr[q][0] = r0 * ch + i0 * sh;  vi[q][0] = i0 * ch - r0 * sh;   // * e^{-i th/2}
        vr[q][1] = r1 * ch - i1 * sh;  vi[q][1] = i1 * ch + r1 * sh;   // * e^{+i th/2}
    };
    auto applyRY = [&](int q, float th) {
        float sh, ch; __sincosf(0.5f * th, &sh, &ch);
        float r0 = vr[q][0], i0 = vi[q][0], r1 = vr[q][1], i1 = vi[q][1];
        vr[q][0] = ch * r0 - sh * r1;  vi[q][0] = ch * i0 - sh * i1;
        vr[q][1] = sh * r0 + ch * r1;  vi[q][1] = sh * i0 + ch * i1;
    };
#pragma unroll
    for (int q = 0; q < 3; ++q) {
        applyRZ(q, t[q * 5 + 0]); applyRY(q, t[q * 5 + 1]); applyRZ(q, t[q * 5 + 2]);
        applyRY(q, t[q * 5 + 3]); applyRZ(q, t[q * 5 + 4]);
    }
    applyRZ(3, t[15]);

    // amplitudes = outer product; apply shared CRZ phases; stage re/im to LDS
    float w01r[4], w01i[4], w23r[4], w23i[4];
#pragma unroll
    for (int i = 0; i < 2; ++i)
#pragma unroll
        for (int j = 0; j < 2; ++j) {
            w01r[i * 2 + j] = vr[0][i] * vr[1][j] - vi[0][i] * vi[1][j];
            w01i[i * 2 + j] = vr[0][i] * vi[1][j] + vi[0][i] * vr[1][j];
            w23r[i * 2 + j] = vr[2][i] * vr[3][j] - vi[2][i] * vi[3][j];
            w23i[i * 2 + j] = vr[2][i] * vi[3][j] + vi[2][i] * vr[3][j];
        }
#pragma unroll
    for (int s = 0; s < 16; ++s) {
        float ar = w01r[s >> 2], ai = w01i[s >> 2];
        float br = w23r[s & 3],  bi = w23i[s & 3];
        float re = ar * br - ai * bi, im = ar * bi + ai * br;
        float pc = s_phc[s], ps = s_phs[s];
        s_re[wv][lane][s] = re * pc - im * ps;
        s_im[wv][lane][s] = re * ps + im * pc;
    }
    __syncthreads();

    // ---------------- per-wave: two 16-row WMMA tiles ----------------
    const int nn   = lane & 15;
    const int koff = (lane < 16) ? 0 : 2;        // A/B K split across half-waves

#pragma unroll
    for (int tI = 0; tI < 2; ++tI) {
        v8f dRe = {}; v8f dIm = {};
        const int srcRow = tI * 16 + nn;
#pragma unroll
        for (int k0 = 0; k0 < 16; k0 += 4) {
            const int kk = k0 + koff;
            v2f a_re, a_im, bop;
            a_re.x = s_re[wv][srcRow][kk];      a_re.y = s_re[wv][srcRow][kk + 1];
            a_im.x = s_im[wv][srcRow][kk];      a_im.y = s_im[wv][srcRow][kk + 1];
            bop.x  = s_bmat[kk * 16 + nn];      bop.y  = s_bmat[(kk + 1) * 16 + nn];
            dRe = __builtin_amdgcn_wmma_f32_16x16x4_f32(false, a_re, false, bop,
                                                        (short)0, dRe, false, false);
            dIm = __builtin_amdgcn_wmma_f32_16x16x4_f32(false, a_im, false, bop,
                                                        (short)0, dIm, false, false);
        }
        // probabilities in C-layout -> padded LDS
        const int prow0 = (lane < 16) ? 0 : 8;
#pragma unroll
        for (int v = 0; v < 8; ++v) {
            float p = dRe[v] * dRe[v] + dIm[v] * dIm[v];
            s_p[wv][prow0 + v][nn] = p;
        }
        __builtin_amdgcn_wave_barrier();
        // signed reductions: 64 outputs/tile, 2 per lane
#pragma unroll
        for (int oo = 0; oo < 2; ++oo) {
            int o    = lane + oo * 32;
            int prow = o >> 2;
            int q    = o & 3;
            float ev = 0.f;
#pragma unroll
            for (int s = 0; s < 16; ++s) {
                float sgn = ((s >> (3 - q)) & 1) ? -1.f : 1.f;
                ev += sgn * s_p[wv][prow][s];
            }
            int rowG = rowBase + wv * 32 + tI * 16 + prow;
            if (rowG < totalRows) out[(size_t)rowG * 4 + q] = ev;
        }
        __builtin_amdgcn_wave_barrier();
    }
}

extern "C" void kernel_launch(void* const* d_in, const int* in_sizes, int n_in,
                              void* d_out, int out_size, void* d_ws, size_t ws_size,
                              hipStream_t stream) {
    const float* X  = (const float*)d_in[0];
    const float* Wt = (const float*)d_in[1];
    float* out      = (float*)d_out;
    int B = in_sizes[0] / (HW * HW);
    int totalRows = B * OX * OY;                 // 1,155,200 for B=128
    int blocks = (totalRows + 255) / 256;
    quanv_kernel<<<blocks, 256, 0, stream>>>(X, Wt, out, totalRows);
}